// SpectralContrastiveLoss_1451698946664
// MI455X (gfx1250) — compile-verified
//
#include <hip/hip_runtime.h>

typedef __attribute__((ext_vector_type(2))) float v2f;
typedef __attribute__((ext_vector_type(8))) float v8f;

#define D_FEAT 512
#define KSPLIT 8   // K-slices per tile == waves per block

// ---------------------------------------------------------------------------
// Zero the 3 scalar accumulators (workspace is poisoned 0xAA before timing).
// ---------------------------------------------------------------------------
__global__ void zero_acc(float* __restrict__ acc) {
  int i = threadIdx.x;
  if (i < 3) acc[i] = 0.0f;
}

// ---------------------------------------------------------------------------
// A = X^T X, B = Y^T Y via V_WMMA_F32_16X16X4_F32.
// One BLOCK owns one 32x32 output tile; its 8 waves each cover a K-slice of
// N/8 samples with a 2x2 grid of 16x16 WMMA accumulators (4-deep WMMA ILP,
// 2 loads per WMMA). K-slice partials combine via ds_add_f32 into a 4 KB LDS
// tile, then plain coalesced global stores -- no global atomics, no A/B
// zero-init pass.
// Operand layout (fp32 16x4 A-matrix): lane = 16*half + l,
//   VGPR0 = Z[k + 2*half][feat], VGPR1 = Z[k + 2*half + 1][feat];
// loads are contiguous across the 16 lanes of each half (feature-major).
// ---------------------------------------------------------------------------
__global__ __launch_bounds__(256) void syrk_wmma(const float* __restrict__ X,
                                                 const float* __restrict__ Y,
                                                 float* __restrict__ A,
                                                 float* __restrict__ B,
                                                 int N) {
  const int D = D_FEAT;
  const int tilesPerDim = D / 32;                    // 16
  const int tilesPerMat = tilesPerDim * tilesPerDim; // 256

  __shared__ float tileAcc[32 * 32];

  int blk  = blockIdx.x;
  int mat  = blk / tilesPerMat;                 // 0 -> X^T X, 1 -> Y^T Y
  int tile = blk % tilesPerMat;
  int tm   = (tile / tilesPerDim) * 32;         // output row base
  int tn   = (tile % tilesPerDim) * 32;         // output col base

  const float* __restrict__ Z   = (mat == 0) ? X : Y;
  float*       __restrict__ Out = (mat == 0) ? A : B;

  int tid  = threadIdx.x;
  int wv   = tid >> 5;                          // K-slice index (0..7)
  int lane = tid & 31;
  int hl   = lane >> 4;                         // half-wave (0/1)
  int l    = lane & 15;

  // Zero the LDS accumulator tile.
  for (int i = tid; i < 1024; i += 256) tileAcc[i] = 0.0f;
  __syncthreads();

  int kChunk = N / KSPLIT;
  int k0 = wv * kChunk;
  int k1 = k0 + kChunk;

  v8f acc00 = {}, acc01 = {}, acc10 = {}, acc11 = {};

#pragma unroll 2
  for (int k = k0; k < k1; k += 4) {
    const float* r0 = Z + (k + 2 * hl) * D;     // sample row k (+2 for hi half)
    const float* r1 = r0 + D;                   // sample row k+1
    v2f a0, a1, b0, b1;
    a0.x = r0[tm + l];        a0.y = r1[tm + l];
    a1.x = r0[tm + 16 + l];   a1.y = r1[tm + 16 + l];
    b0.x = r0[tn + l];        b0.y = r1[tn + l];
    b1.x = r0[tn + 16 + l];   b1.y = r1[tn + 16 + l];

    acc00 = __builtin_amdgcn_wmma_f32_16x16x4_f32(false, a0, false, b0,
                                                  (short)0, acc00, false, false);
    acc01 = __builtin_amdgcn_wmma_f32_16x16x4_f32(false, a0, false, b1,
                                                  (short)0, acc01, false, false);
    acc10 = __builtin_amdgcn_wmma_f32_16x16x4_f32(false, a1, false, b0,
                                                  (short)0, acc10, false, false);
    acc11 = __builtin_amdgcn_wmma_f32_16x16x4_f32(false, a1, false, b1,
                                                  (short)0, acc11, false, false);
  }

  // C/D layout: VGPR r, lanes 0-15 -> M=r, lanes 16-31 -> M=r+8, N=l.
  // Combine the 8 K-slice partials in LDS.
  for (int r = 0; r < 8; ++r) {
    int m = r + 8 * hl;
    atomicAdd(&tileAcc[m * 32 + l],             acc00[r]);
    atomicAdd(&tileAcc[m * 32 + 16 + l],        acc01[r]);
    atomicAdd(&tileAcc[(m + 16) * 32 + l],      acc10[r]);
    atomicAdd(&tileAcc[(m + 16) * 32 + 16 + l], acc11[r]);
  }
  __syncthreads();

  // Coalesced tile writeback: thread t writes elements t, t+256, ...
  for (int i = tid; i < 1024; i += 256) {
    int row = i >> 5;
    int col = i & 31;
    Out[(tm + row) * D + tn + col] = tileAcc[i];
  }
}

// ---------------------------------------------------------------------------
// diag_i = <x_i, y_i>; accumulate sum(diag) into acc[0], sum(diag^2) -> acc[1].
// One wave32 per row; coalesced strided reads; shuffle reduction.
// ---------------------------------------------------------------------------
__global__ __launch_bounds__(256) void diag_dot(const float* __restrict__ X,
                                                const float* __restrict__ Y,
                                                float* __restrict__ acc, int N) {
  const int D = D_FEAT;
  int wave = (blockIdx.x * blockDim.x + threadIdx.x) >> 5;
  int lane = threadIdx.x & 31;
  if (wave >= N) return;
  const float* xr = X + wave * D;
  const float* yr = Y + wave * D;
  float s = 0.0f;
  for (int j = lane; j < D; j += 32) s += xr[j] * yr[j];
  for (int off = 16; off > 0; off >>= 1) s += __shfl_xor(s, off, 32);
  if (lane == 0) {
    atomicAdd(&acc[0], s);
    atomicAdd(&acc[1], s * s);
  }
}

// ---------------------------------------------------------------------------
// acc[2] += <A, B> (elementwise over the two 512x512 Gram matrices).
// ---------------------------------------------------------------------------
__global__ __launch_bounds__(256) void trace_ab(const float* __restrict__ A,
                                                const float* __restrict__ B,
                                                float* __restrict__ acc) {
  const int total = D_FEAT * D_FEAT;
  float s = 0.0f;
  for (int i = blockIdx.x * blockDim.x + threadIdx.x; i < total;
       i += gridDim.x * blockDim.x)
    s += A[i] * B[i];
  int lane = threadIdx.x & 31;
  for (int off = 16; off > 0; off >>= 1) s += __shfl_xor(s, off, 32);
  if (lane == 0) atomicAdd(&acc[2], s);
}

// ---------------------------------------------------------------------------
// loss = (tr(AB) - sum diag^2) / (N(N-1)) - 2 * sum diag / N
// ---------------------------------------------------------------------------
__global__ void finalize(const float* __restrict__ acc, float* __restrict__ out,
                         int N) {
  float tr  = acc[2];
  float sd  = acc[0];
  float sd2 = acc[1];
  float Nf  = (float)N;
  out[0] = (tr - sd2) / (Nf * (Nf - 1.0f)) - 2.0f * sd / Nf;
}

extern "C" void kernel_launch(void* const* d_in, const int* in_sizes, int n_in,
                              void* d_out, int out_size, void* d_ws, size_t ws_size,
                              hipStream_t stream) {
  const float* x = (const float*)d_in[0];
  const float* y = (const float*)d_in[1];
  float* out = (float*)d_out;

  const int D = D_FEAT;
  int N = in_sizes[0] / D;          // 8192

  float* A   = (float*)d_ws;        // 512*512 floats
  float* B   = A + D * D;           // 512*512 floats
  float* acc = B + D * D;           // 3 floats: sum(diag), sum(diag^2), tr(AB)

  zero_acc<<<1, 32, 0, stream>>>(acc);

  int tilesPerMat = (D / 32) * (D / 32);              // 256
  syrk_wmma<<<2 * tilesPerMat, 256, 0, stream>>>(x, y, A, B, N);

  diag_dot<<<(N * 32) / 256, 256, 0, stream>>>(x, y, acc, N);
  trace_ab<<<128, 256, 0, stream>>>(A, B, acc);
  finalize<<<1, 1, 0, stream>>>(acc, out, N);
}